// DualBranchFusion_40845138985155
// MI455X (gfx1250) — compile-verified
//
#include <hip/hip_runtime.h>
#include <hip/hip_bf16.h>

typedef __attribute__((ext_vector_type(2))) float v2f;
typedef __attribute__((ext_vector_type(8))) float v8f;

// ---------------------------------------------------------------------------
// Implicit-GEMM 3x3 conv (stride 1, SAME pad) via V_WMMA_F32_16X16X4_F32.
// Workgroup: 128 threads (4 wave32) -> one row-tile of 64 pixels, all Cout.
// K dimension ordered k = t*CIN + c (t = ky*3+kx) so each K-step of 4 stays
// inside one tap.
//
// LDS layouts are chosen so each lane's WMMA fragment halves are contiguous
// 8-byte pairs (single ds_load_b64 into an even-aligned VGPR pair, zero
// shuffle movs), and all per-step address deltas are compile-time constants
// folded into the DS immediate offset:
//   s_in[(r*66+xi)*CINP + c]            CINP = CIN+2 (bank-conflict pad)
//   s_w [(k>>1)*2*NOUT + n*2 + (k&1)]   row-pair interleaved
// ---------------------------------------------------------------------------
template<int CIN, int NTILE, int RELU>
__global__ __launch_bounds__(128)
void conv3_wmma_kernel(const float* __restrict__ srcA,
                       const float* __restrict__ srcB,
                       const float* __restrict__ wgt,
                       const float* __restrict__ bias,
                       float* __restrict__ out,
                       int coutActual, int Bn, int H, int W)
{
    static_assert(CIN % 4 == 0, "CIN must be a multiple of 4");
    constexpr int NOUT = 16 * NTILE;
    constexpr int CINP = CIN + 2;            // stride pad: 16 lanes -> 16 banks
    __shared__ float s_in[3 * 66 * CINP];    // [(r*66+xi)][c]
    __shared__ float s_w[9 * CIN * NOUT];    // row-pair interleaved

    const int tid    = threadIdx.x;
    const int tilesX = W >> 6;
    const int gid    = blockIdx.x;
    const int tx = gid % tilesX;
    const int y  = (gid / tilesX) % H;
    const int b  = gid / (tilesX * H);
    const int x0 = tx << 6;

    // Stage weights (pair-interleaved for ds_load_b64 on the B fragment).
    for (int idx = tid; idx < 9 * CIN * NOUT; idx += 128) {
        const int k = idx / NOUT;
        const int n = idx - k * NOUT;
        const int c = k % CIN;
        const int t = k / CIN;
        float v = 0.0f;
        if (n < coutActual) v = wgt[(n * CIN + c) * 9 + t];
        s_w[(k >> 1) * (2 * NOUT) + n * 2 + (k & 1)] = v;
    }
    // Stage input halo tile, channel-last. Each thread handles whole channel
    // vectors for a (row, x) position: global reads coalesce across lanes per
    // channel, LDS writes are CIN consecutive floats (b128-mergeable).
    for (int pos = tid; pos < 3 * 66; pos += 128) {
        const int r  = pos / 66;
        const int xi = pos - r * 66;
        const int yy = y + r - 1;
        const int xx = x0 + xi - 1;
        const bool inb = (yy >= 0 && yy < H && xx >= 0 && xx < W);
        float* dst = &s_in[pos * CINP];
#pragma unroll
        for (int c = 0; c < CIN; ++c) {
            float v = 0.0f;
            if (inb) {
                const float* src = srcA;
                int cc = c;
                if (CIN == 32 && c >= 16) { src = srcB; cc = c - 16; }
                v = src[((size_t)(b * 16 + cc) * H + yy) * W + xx];
            }
            dst[c] = v;
        }
    }
    __syncthreads();

    const int lane = tid & 31;
    const int Mi   = lane & 15;       // A row (= local pixel), B/C column
    const int half = lane >> 4;       // K half for A/B fragments
    const int m0   = (tid >> 5) << 4; // wave's pixel base within 64-tile

    // Per-lane bases; every loop step below is a pure immediate DS offset.
    const int aBase = (m0 + Mi) * CINP + half * 2;       // + (ky*66+kx)*CINP + c0
    const int bBase = half * (2 * NOUT) + Mi * 2;        // + k0*NOUT + tN*32

    v8f acc[NTILE];
#pragma unroll
    for (int tN = 0; tN < NTILE; ++tN) {
        v8f z = {0.f, 0.f, 0.f, 0.f, 0.f, 0.f, 0.f, 0.f};
        acc[tN] = z;
    }

#pragma unroll
    for (int t = 0; t < 9; ++t) {
        const int ky = t / 3;
        const int kx = t - ky * 3;
        const int aOff = (ky * 66 + kx) * CINP;
#pragma unroll
        for (int c0 = 0; c0 < CIN; c0 += 4) {
            const v2f a = *(const v2f*)&s_in[aBase + aOff + c0];   // K = k0+2h, +1
            const int k0 = t * CIN + c0;
#pragma unroll
            for (int tN = 0; tN < NTILE; ++tN) {
                const v2f bb = *(const v2f*)&s_w[bBase + k0 * NOUT + tN * 32];
                acc[tN] = __builtin_amdgcn_wmma_f32_16x16x4_f32(
                    false, a, false, bb, (short)0, acc[tN], false, false);
            }
        }
    }

    // C layout: VGPR r -> pixel m0 + r + 8*half, column (out-chan) = Mi.
    // Each lane stores 8 contiguous floats (two 32B-aligned b128 stores).
#pragma unroll
    for (int tN = 0; tN < NTILE; ++tN) {
        const int ch = Mi + tN * 16;
        if (ch < coutActual) {
            const float bv = bias[ch];
            float r0 = acc[tN][0] + bv, r1 = acc[tN][1] + bv;
            float r2 = acc[tN][2] + bv, r3 = acc[tN][3] + bv;
            float r4 = acc[tN][4] + bv, r5 = acc[tN][5] + bv;
            float r6 = acc[tN][6] + bv, r7 = acc[tN][7] + bv;
            if (RELU) {
                r0 = fmaxf(r0, 0.f); r1 = fmaxf(r1, 0.f);
                r2 = fmaxf(r2, 0.f); r3 = fmaxf(r3, 0.f);
                r4 = fmaxf(r4, 0.f); r5 = fmaxf(r5, 0.f);
                r6 = fmaxf(r6, 0.f); r7 = fmaxf(r7, 0.f);
            }
            float* op = &out[((size_t)(b * coutActual + ch) * H + y) * W
                             + x0 + m0 + half * 8];
            *(float4*)(op)     = make_float4(r0, r1, r2, r3);
            *(float4*)(op + 4) = make_float4(r4, r5, r6, r7);
        }
    }
}

// ---------------------------------------------------------------------------
// Attention head: M_E = sigmoid(conv3(t1, w2[1,16,3,3]) + b2); f_Esa = f_E*M_E
// Cout == 1 -> plain VALU dot (0.075 GFLOP, not worth the matrix pipe).
// ---------------------------------------------------------------------------
__global__ __launch_bounds__(256)
void att2_mul_kernel(const float* __restrict__ t1, const float* __restrict__ fE,
                     const float* __restrict__ w2, const float* __restrict__ b2,
                     float* __restrict__ fEsa, int Bn, int H, int W)
{
    __shared__ float sw[144];
    const int tid = threadIdx.x;
    if (tid < 144) sw[tid] = w2[tid];
    __syncthreads();
    const int p = blockIdx.x * 256 + tid;
    const int total = Bn * H * W;
    if (p >= total) return;
    const int x = p % W;
    const int y = (p / W) % H;
    const int b = p / (W * H);
    float acc = b2[0];
    for (int c = 0; c < 16; ++c) {
        const float* base = &t1[(size_t)(b * 16 + c) * H * W];
        const float* wc = &sw[c * 9];
#pragma unroll
        for (int ky = 0; ky < 3; ++ky) {
            const int yy = y + ky - 1;
            if (yy < 0 || yy >= H) continue;
#pragma unroll
            for (int kx = 0; kx < 3; ++kx) {
                const int xx = x + kx - 1;
                if (xx < 0 || xx >= W) continue;
                acc += base[(size_t)yy * W + xx] * wc[ky * 3 + kx];
            }
        }
    }
    const float M = 1.0f / (1.0f + __expf(-acc));
    const size_t hw = (size_t)H * W;
    const size_t pix = (size_t)y * W + x;
#pragma unroll
    for (int c = 0; c < 16; ++c) {
        const size_t o = ((size_t)b * 16 + c) * hw + pix;
        fEsa[o] = fE[o] * M;
    }
}

// ---------------------------------------------------------------------------
// Modulated deformable conv (torchvision semantics): per-pixel bilinear
// gather (valid-masked, clipped) * sigmoid(mask), contracted over (c,tap)
// with dcb_w, + bias, ReLU. Gather-bound -> VALU kernel, weights in LDS.
// om buffer: ch 0..17 raw offsets (y,x per tap), ch 18..26 pre-sigmoid mask.
// ---------------------------------------------------------------------------
__global__ __launch_bounds__(256)
void deform_conv_kernel(const float* __restrict__ x, const float* __restrict__ om,
                        const float* __restrict__ wgt, const float* __restrict__ bias,
                        float* __restrict__ out, int Bn, int H, int W)
{
    __shared__ float sw[16 * 16 * 9];   // [o][c][t]
    const int tid = threadIdx.x;
    for (int i = tid; i < 2304; i += 256) sw[i] = wgt[i];
    __syncthreads();
    const int p = blockIdx.x * 256 + tid;
    const int total = Bn * H * W;
    if (p >= total) return;
    const int xw = p % W;
    const int y  = (p / W) % H;
    const int b  = p / (W * H);
    const size_t hw = (size_t)H * W;

    float acc[16];
#pragma unroll
    for (int o = 0; o < 16; ++o) acc[o] = bias[o];

    const float* omb = om + (size_t)b * 27 * hw + (size_t)y * W + xw;
    const float* xb  = x + (size_t)b * 16 * hw;

    for (int t = 0; t < 9; ++t) {
        const int ky = t / 3;
        const int kx = t - ky * 3;
        const float offy = omb[(size_t)(2 * t) * hw];
        const float offx = omb[(size_t)(2 * t + 1) * hw];
        float mval = omb[(size_t)(18 + t) * hw];
        mval = 1.0f / (1.0f + __expf(-mval));
        const float py = (float)(y + ky - 1) + offy;
        const float px = (float)(xw + kx - 1) + offx;
        const float y0f = floorf(py);
        const float x0f = floorf(px);
        const float dy = py - y0f;
        const float dx = px - x0f;
        const int y0 = (int)y0f, x0 = (int)x0f;
        const int y1 = y0 + 1,   x1 = x0 + 1;
        const bool vy0 = (y0 >= 0) && (y0 < H);
        const bool vy1 = (y1 >= 0) && (y1 < H);
        const bool vx0 = (x0 >= 0) && (x0 < W);
        const bool vx1 = (x1 >= 0) && (x1 < W);
        const int y0c = min(max(y0, 0), H - 1);
        const int y1c = min(max(y1, 0), H - 1);
        const int x0c = min(max(x0, 0), W - 1);
        const int x1c = min(max(x1, 0), W - 1);
        const float w00 = (vy0 && vx0) ? (1.f - dy) * (1.f - dx) * mval : 0.f;
        const float w01 = (vy0 && vx1) ? (1.f - dy) * dx * mval : 0.f;
        const float w10 = (vy1 && vx0) ? dy * (1.f - dx) * mval : 0.f;
        const float w11 = (vy1 && vx1) ? dy * dx * mval : 0.f;
        const size_t i00 = (size_t)y0c * W + x0c;
        const size_t i01 = (size_t)y0c * W + x1c;
        const size_t i10 = (size_t)y1c * W + x0c;
        const size_t i11 = (size_t)y1c * W + x1c;
        for (int c = 0; c < 16; ++c) {
            const float* img = xb + (size_t)c * hw;
            const float s = img[i00] * w00 + img[i01] * w01
                          + img[i10] * w10 + img[i11] * w11;
#pragma unroll
            for (int o = 0; o < 16; ++o)
                acc[o] += s * sw[o * 144 + c * 9 + t];
        }
    }
#pragma unroll
    for (int o = 0; o < 16; ++o)
        out[((size_t)(b * 16 + o) * H + y) * W + xw] = fmaxf(acc[o], 0.0f);
}

// ---------------------------------------------------------------------------
extern "C" void kernel_launch(void* const* d_in, const int* in_sizes, int n_in,
                              void* d_out, int out_size, void* d_ws, size_t ws_size,
                              hipStream_t stream) {
    (void)in_sizes; (void)n_in; (void)out_size; (void)ws_size;
    const float* f_E = (const float*)d_in[0];
    const float* f_F = (const float*)d_in[1];
    const float* aw1 = (const float*)d_in[2];
    const float* ab1 = (const float*)d_in[3];
    const float* aw2 = (const float*)d_in[4];
    const float* ab2 = (const float*)d_in[5];
    const float* rw  = (const float*)d_in[6];
    const float* rb  = (const float*)d_in[7];
    const float* ow1 = (const float*)d_in[8];
    const float* ob1 = (const float*)d_in[9];
    const float* ow2 = (const float*)d_in[10];
    const float* ob2 = (const float*)d_in[11];
    const float* ow3 = (const float*)d_in[12];
    const float* ob3 = (const float*)d_in[13];
    const float* dw  = (const float*)d_in[14];
    const float* db  = (const float*)d_in[15];
    const float* drw = (const float*)d_in[16];
    const float* drb = (const float*)d_in[17];
    const float* fw  = (const float*)d_in[18];
    const float* fb  = (const float*)d_in[19];

    const int B = 4, H = 256, W = 256;
    const size_t S = (size_t)B * 16 * H * W;   // one 16-channel fp32 buffer

    float* ws   = (float*)d_ws;
    float* buf0 = ws;            // t1 -> om1 -> f_Edc
    float* buf1 = ws + S;        // f_Esa -> om2 -> f_EFdc
    float* buf2 = ws + 2 * S;    // f_EFsa (live to the end)
    float* om   = ws + 3 * S;    // 27-channel offset/mask conv output

    dim3 cgrid((unsigned)(B * H * (W / 64))), cblk(128);
    const int total = B * H * W;
    dim3 pgrid((unsigned)((total + 255) / 256)), pblk(256);

    // t1 = relu(conv(concat(f_E,f_F), sab_att_w1))
    conv3_wmma_kernel<32, 1, 1><<<cgrid, cblk, 0, stream>>>(f_E, f_F, aw1, ab1, buf0, 16, B, H, W);
    // f_Esa = f_E * sigmoid(conv(t1, sab_att_w2))
    att2_mul_kernel<<<pgrid, pblk, 0, stream>>>(buf0, f_E, aw2, ab2, buf1, B, H, W);
    // f_EFsa = relu(conv(concat(f_Esa,f_F), sab_ref_w))
    conv3_wmma_kernel<32, 1, 1><<<cgrid, cblk, 0, stream>>>(buf1, f_F, rw, rb, buf2, 16, B, H, W);
    // om1 = relu(conv(concat(f_E,f_F), om_w1))      (t1 dead -> buf0)
    conv3_wmma_kernel<32, 1, 1><<<cgrid, cblk, 0, stream>>>(f_E, f_F, ow1, ob1, buf0, 16, B, H, W);
    // om2 = relu(conv(om1, om_w2))                  (f_Esa dead -> buf1)
    conv3_wmma_kernel<16, 1, 1><<<cgrid, cblk, 0, stream>>>(buf0, nullptr, ow2, ob2, buf1, 16, B, H, W);
    // om = conv(om2, om_w3)   (27 channels, no activation; sigmoid in deform)
    conv3_wmma_kernel<16, 2, 0><<<cgrid, cblk, 0, stream>>>(buf1, nullptr, ow3, ob3, om, 27, B, H, W);
    // f_Edc = relu(deform_conv(f_E, offsets, mask, dcb_w))   (-> buf0)
    deform_conv_kernel<<<pgrid, pblk, 0, stream>>>(f_E, om, dw, db, buf0, B, H, W);
    // f_EFdc = relu(conv(concat(f_Edc,f_F), dcbref_w))       (-> buf1)
    conv3_wmma_kernel<32, 1, 1><<<cgrid, cblk, 0, stream>>>(buf0, f_F, drw, drb, buf1, 16, B, H, W);
    // f_EF = relu(conv(concat(f_EFsa,f_EFdc), fin_w))
    conv3_wmma_kernel<32, 1, 1><<<cgrid, cblk, 0, stream>>>(buf2, buf1, fw, fb, (float*)d_out, 16, B, H, W);
}